// PointnetFPModule_28922309771871
// MI455X (gfx1250) — compile-verified
//
#include <hip/hip_runtime.h>

typedef __attribute__((ext_vector_type(2))) float v2f;
typedef __attribute__((ext_vector_type(8))) float v8f;

static constexpr int B_  = 16;    // batch
static constexpr int N_  = 4096;  // unknown points
static constexpr int M_  = 1024;  // known points
static constexpr int C1_ = 128;   // unknow_feats channels
static constexpr int C_  = 256;   // mlp width / known_feats channels
static constexpr float BN_EPS = 1e-5f;

// gfx1250 async global->LDS copy path (ASYNCcnt-tracked), guarded so a
// toolchain without these builtin names still compiles via the plain path.
#if defined(__has_builtin)
#if __has_builtin(__builtin_amdgcn_global_load_async_to_lds_b32) && \
    __has_builtin(__builtin_amdgcn_s_wait_asynccnt)
#define HAVE_ASYNC_LDS 1
#endif
#endif
#ifndef HAVE_ASYNC_LDS
#define HAVE_ASYNC_LDS 0
#endif

// ---------------------------------------------------------------------------
// Zero the 3 BN stats regions (sum[256], sumsq[256] each, stride 512 floats).
// Must run every launch: the harness graph-replays without re-poisoning.
// ---------------------------------------------------------------------------
__global__ void zero_stats_kernel(float* __restrict__ stats) {
  for (int i = threadIdx.x; i < 3 * 512; i += blockDim.x) stats[i] = 0.0f;
}

// ---------------------------------------------------------------------------
// three_nn + inverse-distance weights. One block = 256 unknown points of one
// batch; known points staged in LDS (3 * 1024 * 4B = 12 KB) via async copies.
// ---------------------------------------------------------------------------
__global__ __launch_bounds__(256) void three_nn_kernel(
    const float* __restrict__ unknown,  // (B, n, 3)
    const float* __restrict__ known,    // (B, m, 3)
    int* __restrict__ idx,              // (B, n, 3)
    float* __restrict__ wgt) {          // (B, n, 3)
  __shared__ float kx[M_], ky[M_], kz[M_];
  const int b = blockIdx.y;
  const float* kb = known + (size_t)b * M_ * 3;

#if HAVE_ASYNC_LDS
  typedef __attribute__((address_space(1))) int glb_i32;  // prints as __device__ int*
  typedef __attribute__((address_space(3))) int lds_i32;  // prints as __shared__ int*
  for (int t = threadIdx.x; t < M_; t += blockDim.x) {
    __builtin_amdgcn_global_load_async_to_lds_b32(
        (glb_i32*)(kb + t * 3 + 0), (lds_i32*)&kx[t], 0, 0);
    __builtin_amdgcn_global_load_async_to_lds_b32(
        (glb_i32*)(kb + t * 3 + 1), (lds_i32*)&ky[t], 0, 0);
    __builtin_amdgcn_global_load_async_to_lds_b32(
        (glb_i32*)(kb + t * 3 + 2), (lds_i32*)&kz[t], 0, 0);
  }
  __builtin_amdgcn_s_wait_asynccnt(0);
  __syncthreads();
#else
  for (int t = threadIdx.x; t < M_; t += blockDim.x) {
    kx[t] = kb[t * 3 + 0];
    ky[t] = kb[t * 3 + 1];
    kz[t] = kb[t * 3 + 2];
  }
  __syncthreads();
#endif

  const int i = blockIdx.x * blockDim.x + threadIdx.x;
  const float* up = unknown + ((size_t)b * N_ + i) * 3;
  const float px = up[0], py = up[1], pz = up[2];

  float d0 = 3.4e38f, d1 = 3.4e38f, d2 = 3.4e38f;
  int i0 = 0, i1 = 0, i2 = 0;
  for (int j = 0; j < M_; ++j) {
    const float dx = px - kx[j], dy = py - ky[j], dz = pz - kz[j];
    const float d = dx * dx + dy * dy + dz * dz;
    if (d < d0)      { d2 = d1; i2 = i1; d1 = d0; i1 = i0; d0 = d; i0 = j; }
    else if (d < d1) { d2 = d1; i2 = i1; d1 = d;  i1 = j; }
    else if (d < d2) { d2 = d;  i2 = j; }
  }
  const float r0 = 1.0f / (d0 + 1e-8f);
  const float r1 = 1.0f / (d1 + 1e-8f);
  const float r2 = 1.0f / (d2 + 1e-8f);
  const float rn = 1.0f / (r0 + r1 + r2);
  const size_t base = ((size_t)b * N_ + i) * 3;
  idx[base + 0] = i0; idx[base + 1] = i1; idx[base + 2] = i2;
  wgt[base + 0] = r0 * rn; wgt[base + 1] = r1 * rn; wgt[base + 2] = r2 * rn;
}

// ---------------------------------------------------------------------------
// three_interpolate: out(b,c,j) = sum_k kf(b,c,idx_k) * w_k.
// Thread = column j, loop over channels -> coalesced stores, L2-friendly
// gathers (each (b,c) row of known_feats is only 4 KB).
// ---------------------------------------------------------------------------
__global__ __launch_bounds__(256) void interp_kernel(
    const float* __restrict__ kf,   // (B, C_, M_)
    const int* __restrict__ idx,
    const float* __restrict__ wgt,
    float* __restrict__ out) {      // (B, C_, N_)
  const int b = blockIdx.y;
  const int j = blockIdx.x * blockDim.x + threadIdx.x;
  const size_t base = ((size_t)b * N_ + j) * 3;
  const int   i0 = idx[base + 0], i1 = idx[base + 1], i2 = idx[base + 2];
  const float w0 = wgt[base + 0], w1 = wgt[base + 1], w2 = wgt[base + 2];
  const float* kb = kf + (size_t)b * C_ * M_;
  float* ob = out + (size_t)b * C_ * N_;
  for (int c = 0; c < C_; ++c) {
    const float* row = kb + (size_t)c * M_;
    ob[(size_t)c * N_ + j] = row[i0] * w0 + row[i1] * w1 + row[i2] * w2;
  }
}

// ---------------------------------------------------------------------------
// Pure WMMA GEMM: Y(b) = W(256 x CIN) @ X(b)(CIN x N) + bias.
// One wave owns one 16-column slab and all 16 row-tiles (v8f acc[16] = 128
// accumulator VGPRs), K-steps of 4 with V_WMMA_F32_16X16X4_F32.
//   A (16x4): lane<16 -> {K+0,K+1}, lane>=16 -> {K+2,K+3} of row (lane&15)
//   B (4x16): same K split, column = ct*16 + (lane&15)
//   C/D     : vgpr r -> row rt*16 + r (+8 for lanes 16..31), col as above
// No LDS / barriers / atomics: BN stats are a separate L2-resident pass.
// ---------------------------------------------------------------------------
template <int CIN>
__global__ __launch_bounds__(256) void gemm_wmma_kernel(
    const float* __restrict__ X,     // (B, CIN, N_)
    const float* __restrict__ W,     // (C_, CIN) row-major
    const float* __restrict__ bias,  // (C_)
    float* __restrict__ Y) {         // (B, C_, N_)
  const int b     = blockIdx.z;
  const int wave  = threadIdx.x >> 5;
  const int lane  = threadIdx.x & 31;
  const int ct    = blockIdx.x * 8 + wave;       // 16-col tile index
  const int col   = ct * 16 + (lane & 15);
  const int khalf = (lane >> 4) << 1;            // 0 or 2
  const float* Xb = X + (size_t)b * CIN * N_;

  v8f acc[16];
#pragma unroll
  for (int rt = 0; rt < 16; ++rt) acc[rt] = v8f{};

  for (int k = 0; k < CIN; k += 4) {
    v2f bb;
    bb.x = Xb[(size_t)(k + khalf + 0) * N_ + col];
    bb.y = Xb[(size_t)(k + khalf + 1) * N_ + col];
#pragma unroll
    for (int rt = 0; rt < 16; ++rt) {
      const float* wp = W + (size_t)(rt * 16 + (lane & 15)) * CIN + k + khalf;
      v2f aa = *(const v2f*)wp;  // 8B aligned: even float offset
      acc[rt] = __builtin_amdgcn_wmma_f32_16x16x4_f32(
          false, aa, false, bb, (short)0, acc[rt], false, false);
    }
  }

  float* Yb = Y + (size_t)b * C_ * N_;
  const int mofs = (lane < 16) ? 0 : 8;
#pragma unroll
  for (int rt = 0; rt < 16; ++rt) {
#pragma unroll
    for (int r = 0; r < 8; ++r) {
      const int row = rt * 16 + mofs + r;
      Yb[(size_t)row * N_ + col] = acc[rt][r] + bias[row];
    }
  }
}

// ---------------------------------------------------------------------------
// Per-channel sum / sumsq over (B, N). One block per (channel, batch) row:
// float4-coalesced reads out of L2 (Y was just written, 67 MB << 192 MB L2),
// per-wave butterfly + tiny LDS combine, two global atomics per block.
// ---------------------------------------------------------------------------
__global__ __launch_bounds__(256) void bn_stats_kernel(
    const float* __restrict__ Y, float* __restrict__ stats) {
  const int c = blockIdx.x;   // channel
  const int b = blockIdx.y;   // batch
  const float4* row = (const float4*)(Y + ((size_t)b * C_ + c) * N_);
  float s = 0.0f, q = 0.0f;
#pragma unroll
  for (int it = 0; it < N_ / 4 / 256; ++it) {
    const float4 v = row[it * 256 + threadIdx.x];
    s += v.x + v.y + v.z + v.w;
    q += v.x * v.x + v.y * v.y + v.z * v.z + v.w * v.w;
  }
#pragma unroll
  for (int o = 1; o < 32; o <<= 1) {
    s += __shfl_xor(s, o, 32);
    q += __shfl_xor(q, o, 32);
  }
  __shared__ float wsum[8], wsq[8];
  const int wave = threadIdx.x >> 5, lane = threadIdx.x & 31;
  if (lane == 0) { wsum[wave] = s; wsq[wave] = q; }
  __syncthreads();
  if (threadIdx.x == 0) {
    float ts = 0.0f, tq = 0.0f;
#pragma unroll
    for (int w = 0; w < 8; ++w) { ts += wsum[w]; tq += wsq[w]; }
    atomicAdd(&stats[c], ts);
    atomicAdd(&stats[256 + c], tq);
  }
}

// ---------------------------------------------------------------------------
// Convert (sum, sumsq) -> (mean, rstd) in place. One block of 256 threads.
// ---------------------------------------------------------------------------
__global__ void bn_finalize_kernel(float* __restrict__ stats) {
  const int c = threadIdx.x;
  const float inv = 1.0f / (float)((size_t)B_ * N_);
  const float mean = stats[c] * inv;
  const float var  = stats[256 + c] * inv - mean * mean;
  stats[c]       = mean;
  stats[256 + c] = rsqrtf(var + BN_EPS);
}

// ---------------------------------------------------------------------------
// out = [relu]( (Y - mean) * rstd * gamma + beta [+ skip] )
// ---------------------------------------------------------------------------
template <bool RELU>
__global__ __launch_bounds__(256) void bn_apply_kernel(
    const float* __restrict__ Y, const float* __restrict__ stats,
    const float* __restrict__ gamma, const float* __restrict__ beta,
    const float* __restrict__ skip, float* __restrict__ out) {
  const size_t total = (size_t)B_ * C_ * N_;
  for (size_t t = (size_t)blockIdx.x * blockDim.x + threadIdx.x; t < total;
       t += (size_t)gridDim.x * blockDim.x) {
    const int c = (int)((t / N_) % C_);
    float v = (Y[t] - stats[c]) * stats[256 + c] * gamma[c] + beta[c];
    if (skip) v += skip[t];
    if (RELU) v = fmaxf(v, 0.0f);
    out[t] = v;
  }
}

// ---------------------------------------------------------------------------
extern "C" void kernel_launch(void* const* d_in, const int* in_sizes, int n_in,
                              void* d_out, int out_size, void* d_ws, size_t ws_size,
                              hipStream_t stream) {
  const float* unknown = (const float*)d_in[0];
  const float* known   = (const float*)d_in[1];
  const float* uf      = (const float*)d_in[2];   // (B, 128, n)
  const float* kf      = (const float*)d_in[3];   // (B, 256, m)
  const float* Wu = (const float*)d_in[4];  const float* bu    = (const float*)d_in[5];
  const float* gu = (const float*)d_in[6];  const float* betau = (const float*)d_in[7];
  const float* W1 = (const float*)d_in[8];  const float* b1    = (const float*)d_in[9];
  const float* g1 = (const float*)d_in[10]; const float* beta1 = (const float*)d_in[11];
  const float* W2 = (const float*)d_in[12]; const float* b2    = (const float*)d_in[13];
  const float* g2 = (const float*)d_in[14]; const float* beta2 = (const float*)d_in[15];
  float* out = (float*)d_out;

  char* ws = (char*)d_ws;
  const size_t idx_bytes = (size_t)B_ * N_ * 3 * sizeof(int);   // 786432
  const size_t buf_bytes = (size_t)B_ * C_ * N_ * sizeof(float);// 67108864
  float* stats = (float*)ws;                                     // 3 * 512 f32
  int*   idx   = (int*)(ws + 8192);
  float* wgt   = (float*)(ws + 8192 + idx_bytes);
  float* bufA  = (float*)(ws + 8192 + 2 * idx_bytes);            // interp / x / x1
  float* bufB  = (float*)(ws + 8192 + 2 * idx_bytes + buf_bytes);// y0 / y1 / y2

  const dim3 blk(256);
  const dim3 grid_pts(N_ / 256, B_);
  const dim3 grid_gemm(N_ / 128, 1, B_);  // 8 waves/block * 16 cols = 128 cols
  const dim3 grid_stat(C_, B_);           // one block per (channel, batch) row

  zero_stats_kernel<<<1, blk, 0, stream>>>(stats);
  three_nn_kernel<<<grid_pts, blk, 0, stream>>>(unknown, known, idx, wgt);
  interp_kernel<<<grid_pts, blk, 0, stream>>>(kf, idx, wgt, bufA);

  // skip branch: y0 = Wu @ uf + bu ; x = interp + BN(y0)
  gemm_wmma_kernel<C1_><<<grid_gemm, blk, 0, stream>>>(uf, Wu, bu, bufB);
  bn_stats_kernel<<<grid_stat, blk, 0, stream>>>(bufB, stats);
  bn_finalize_kernel<<<1, blk, 0, stream>>>(stats);
  bn_apply_kernel<false><<<2048, blk, 0, stream>>>(bufB, stats, gu, betau, bufA, bufA);

  // x1 = relu(BN(W1 @ x + b1))
  gemm_wmma_kernel<C_><<<grid_gemm, blk, 0, stream>>>(bufA, W1, b1, bufB);
  bn_stats_kernel<<<grid_stat, blk, 0, stream>>>(bufB, stats + 512);
  bn_finalize_kernel<<<1, blk, 0, stream>>>(stats + 512);
  bn_apply_kernel<true><<<2048, blk, 0, stream>>>(bufB, stats + 512, g1, beta1, nullptr, bufA);

  // out = relu(BN(W2 @ x1 + b2))
  gemm_wmma_kernel<C_><<<grid_gemm, blk, 0, stream>>>(bufA, W2, b2, bufB);
  bn_stats_kernel<<<grid_stat, blk, 0, stream>>>(bufB, stats + 1024);
  bn_finalize_kernel<<<1, blk, 0, stream>>>(stats + 1024);
  bn_apply_kernel<true><<<2048, blk, 0, stream>>>(bufB, stats + 1024, g2, beta2, nullptr, out);
}